// RamaWholePoseScoringModule_81003083203296
// MI455X (gfx1250) — compile-verified
//
#include <hip/hip_runtime.h>
#include <math.h>

// Problem constants (match the reference).
#define NPOSE 512
#define NBLK  512
#define NATM  32
#define NBTC  40
#define NCONN 3
#define NTABC 40
#define NG    36
#define NAPB  (NBLK * NATM)   // atoms per pose

#ifndef __has_builtin
#define __has_builtin(x) 0
#endif

#if defined(__HIP_DEVICE_COMPILE__) && __has_builtin(__builtin_amdgcn_global_load_async_to_lds_b128)
#define HAVE_ASYNC_LDS 1
#else
#define HAVE_ASYNC_LDS 0
#endif

// B128 async copy operates on int4 vectors; builtin expects typed pointers:
//   arg0: int4 addrspace(1)* (global source, prints as "__device__")
//   arg1: int4 addrspace(3)* (LDS destination)
typedef int v4i __attribute__((ext_vector_type(4)));
typedef __attribute__((address_space(1))) v4i as1_v4i;
typedef __attribute__((address_space(3))) v4i as3_v4i;

struct V3 { float x, y, z; };
static __device__ __forceinline__ V3 v3sub(V3 a, V3 b) { return {a.x - b.x, a.y - b.y, a.z - b.z}; }
static __device__ __forceinline__ V3 v3cross(V3 a, V3 b) {
  return {a.y * b.z - a.z * b.y, a.z * b.x - a.x * b.z, a.x * b.y - a.y * b.x};
}
static __device__ __forceinline__ float v3dot(V3 a, V3 b) { return a.x * b.x + a.y * b.y + a.z * b.z; }

static __device__ __forceinline__ float dihedral4(const V3 q[4]) {
  V3 b1 = v3sub(q[1], q[0]);
  V3 b2 = v3sub(q[2], q[1]);
  V3 b3 = v3sub(q[3], q[2]);
  V3 n1 = v3cross(b1, b2);
  V3 n2 = v3cross(b2, b3);
  float inv = 1.0f / (sqrtf(v3dot(b2, b2)) + 1e-12f);
  V3 b2n = {b2.x * inv, b2.y * inv, b2.z * inv};
  V3 m1 = v3cross(n1, b2n);
  float xx = v3dot(n1, n2);
  float yy = v3dot(m1, n2);
  return atan2f(yy, xx);
}

static __device__ __forceinline__ void crweights(float f, float w[4]) {
  float f2 = f * f, f3 = f2 * f;
  w[0] = 0.5f * (-f3 + 2.0f * f2 - f);
  w[1] = 0.5f * (3.0f * f3 - 5.0f * f2 + 2.0f);
  w[2] = 0.5f * (-3.0f * f3 + 4.0f * f2 + f);
  w[3] = 0.5f * (f3 - f2);
}

// Cooperative 16B-granularity stage of a global region into LDS.
// Async path: global_load_async_to_lds_b128 (tracked by ASYNCcnt).
// Fallback: plain uint4 copy (DScnt), resolved by __syncthreads.
static __device__ __forceinline__ void stage16(const void* g, void* s, int bytes) {
  const int t  = (int)threadIdx.x;
  const int nt = (int)blockDim.x;
#if HAVE_ASYNC_LDS
  const char* gc = (const char*)g;
  char* sc = (char*)s;
  for (int i = t * 16; i < bytes; i += nt * 16) {
    // AS1 pointer value == flat value; AS3 pointer == low 32 bits of flat
    // address (ISA 10.2: LDS uses the low 32 address bits), so int-casts are exact.
    __builtin_amdgcn_global_load_async_to_lds_b128(
        (as1_v4i*)(unsigned long long)(gc + i),
        (as3_v4i*)(unsigned int)(unsigned long long)(sc + i),
        0, 0);
  }
#else
  const uint4* gq = (const uint4*)g;
  uint4* sq = (uint4*)s;
  const int n = bytes >> 4;
  for (int i = t; i < n; i += nt) sq[i] = gq[i];
#endif
}

static __device__ __forceinline__ void wait_stage_and_barrier() {
#if HAVE_ASYNC_LDS
#if __has_builtin(__builtin_amdgcn_s_wait_asynccnt)
  __builtin_amdgcn_s_wait_asynccnt(0);
#else
  asm volatile("s_wait_asynccnt 0" ::: "memory");
#endif
#endif
  __syncthreads();
}

__global__ __launch_bounds__(NBLK, 2)
void rama_pose_score_kernel(
    const float* __restrict__ coords,   // (P, NAPB, 3)
    const int*   __restrict__ g_off,    // (P, B)
    const int*   __restrict__ g_bt,     // (P, B)
    const int*   __restrict__ g_inter,  // (P, B, NC, 2)
    const int*   __restrict__ g_down,   // (NBT, NC, A)
    const int*   __restrict__ g_rtab,   // (NBT, 2)
    const int*   __restrict__ g_ucon,   // (NBT)
    const int*   __restrict__ g_ispro,  // (NBT)
    const int*   __restrict__ g_tor,    // (NBT, 8, 3)
    const float* __restrict__ g_tables, // (NTAB, G, G)
    const float* __restrict__ g_tp,     // (NTAB, 2, 2)
    float*       __restrict__ out)      // (P)
{
  __shared__ __align__(16) int   s_off[NBLK];
  __shared__ __align__(16) int   s_bt[NBLK];
  __shared__ __align__(16) int   s_inter[NBLK * NCONN * 2];
  __shared__ __align__(16) int   s_down[NBTC * NCONN * NATM];
  __shared__ __align__(16) int   s_rtab[NBTC * 2];
  __shared__ __align__(16) int   s_ucon[NBTC];
  __shared__ __align__(16) int   s_ispro[NBTC];
  __shared__ __align__(16) int   s_tor[NBTC * 8 * 3];
  __shared__ __align__(16) float s_tp[NTABC * 4];
  __shared__ float s_part[NBLK / 32];

  const int p = (int)blockIdx.x;
  const int b = (int)threadIdx.x;

  // Stage per-pose metadata and small block-type tables into LDS (async).
  stage16(g_off   + (size_t)p * NBLK,             s_off,   NBLK * 4);
  stage16(g_bt    + (size_t)p * NBLK,             s_bt,    NBLK * 4);
  stage16(g_inter + (size_t)p * NBLK * NCONN * 2, s_inter, NBLK * NCONN * 2 * 4);
  stage16(g_down,  s_down,  NBTC * NCONN * NATM * 4);
  stage16(g_rtab,  s_rtab,  NBTC * 2 * 4);
  stage16(g_ucon,  s_ucon,  NBTC * 4);
  stage16(g_ispro, s_ispro, NBTC * 4);
  stage16(g_tor,   s_tor,   NBTC * 8 * 3 * 4);
  stage16(g_tp,    s_tp,    NTABC * 4 * 4);
  wait_stage_and_barrier();

  const int bt    = s_bt[b];
  const int myoff = s_off[b];
  const float* pc = coords + (size_t)p * NAPB * 3;

  float ang[2];
  bool  dok[2];
#pragma unroll
  for (int d = 0; d < 2; ++d) {
    V3 q[4];
    bool allok = true;
#pragma unroll
    for (int t = 0; t < 4; ++t) {
      const int base = (bt * 8 + d * 4 + t) * 3;
      const int atom = s_tor[base + 0];
      const int conn = s_tor[base + 1];
      const int nbnd = s_tor[base + 2];
      int gidx;
      if (atom >= 0) {
        gidx = myoff + atom;  // local atom
      } else {
        const int sc   = conn < 0 ? 0 : conn;
        const int oblk = s_inter[(b * NCONN + sc) * 2 + 0];
        const int ocon = s_inter[(b * NCONN + sc) * 2 + 1];
        allok = allok && (conn >= 0) && (oblk >= 0);
        const int ob  = oblk < 0 ? 0 : oblk;   // clipped, always in-range
        const int oc  = ocon < 0 ? 0 : ocon;
        const int obt = s_bt[ob];
        gidx = s_off[ob] + s_down[(obt * NCONN + oc) * NATM + nbnd];
      }
      const float* cp = pc + (size_t)gidx * 3;
      q[t] = {cp[0], cp[1], cp[2]};
    }
    dok[d] = allok;
    ang[d] = dihedral4(q);  // garbage-in/masked-out matches reference
  }

  // Table selection via downstream neighbor's is_pro flag.
  const int  uc    = s_ucon[bt];
  const int  nb    = s_inter[(b * NCONN + uc) * 2 + 0];
  const bool nb_ok = nb >= 0;
  const int  nbt2  = s_bt[nb < 0 ? 0 : nb];
  const int  tab   = s_rtab[bt * 2 + s_ispro[nbt2]];

  const float u0  = (ang[0] - s_tp[tab * 4 + 0]) / s_tp[tab * 4 + 2];
  const float u1  = (ang[1] - s_tp[tab * 4 + 1]) / s_tp[tab * 4 + 3];
  const float fl0 = floorf(u0), fl1 = floorf(u1);
  const float f0  = u0 - fl0,   f1  = u1 - fl1;
  const long long i0 = (long long)(int)fl0;
  const long long i1 = (long long)(int)fl1;

  int ii[4], jj[4];
#pragma unroll
  for (int k = 0; k < 4; ++k) {
    int v = (int)((i0 + (k - 1)) % NG); if (v < 0) v += NG;
    ii[k] = v;
    int w = (int)((i1 + (k - 1)) % NG); if (w < 0) w += NG;
    jj[k] = w;
  }
  float wphi[4], wpsi[4];
  crweights(f0, wphi);
  crweights(f1, wpsi);

  // 16 table gathers from L2-resident rama tables (207 KB total).
  const float* tbp = g_tables + (size_t)tab * NG * NG;
  float e = 0.0f;
#pragma unroll
  for (int a = 0; a < 4; ++a) {
    const float* rowp = tbp + ii[a] * NG;
    float acc = 0.0f;
#pragma unroll
    for (int c = 0; c < 4; ++c) acc = fmaf(wpsi[c], rowp[jj[c]], acc);
    e = fmaf(wphi[a], acc, e);
  }

  float val = (dok[0] && dok[1] && nb_ok) ? e : 0.0f;

  // wave32 butterfly reduction, then cross-wave via LDS (deterministic order).
#pragma unroll
  for (int o = 16; o > 0; o >>= 1) val += __shfl_xor(val, o, 32);
  const int lane = b & 31, wv = b >> 5;
  if (lane == 0) s_part[wv] = val;
  __syncthreads();
  if (wv == 0) {
    float t = (lane < (NBLK / 32)) ? s_part[lane] : 0.0f;
#pragma unroll
    for (int o = 8; o > 0; o >>= 1) t += __shfl_xor(t, o, 32);
    if (lane == 0) out[p] = t;
  }
}

extern "C" void kernel_launch(void* const* d_in, const int* in_sizes, int n_in,
                              void* d_out, int out_size, void* d_ws, size_t ws_size,
                              hipStream_t stream) {
  (void)in_sizes; (void)n_in; (void)d_ws; (void)ws_size;
  const float* coords  = (const float*)d_in[0];
  const int*   g_off   = (const int*)d_in[1];
  const int*   g_bt    = (const int*)d_in[2];
  const int*   g_inter = (const int*)d_in[3];
  const int*   g_down  = (const int*)d_in[4];
  const int*   g_rtab  = (const int*)d_in[5];
  const int*   g_ucon  = (const int*)d_in[6];
  const int*   g_ispro = (const int*)d_in[7];
  const int*   g_tor   = (const int*)d_in[8];
  const float* g_tab   = (const float*)d_in[9];
  const float* g_tp    = (const float*)d_in[10];
  float* out = (float*)d_out;

  const int nP = out_size;  // one workgroup (16 wave32s) per pose
  rama_pose_score_kernel<<<nP, NBLK, 0, stream>>>(
      coords, g_off, g_bt, g_inter, g_down, g_rtab, g_ucon, g_ispro,
      g_tor, g_tab, g_tp, out);
}